// Encoder_45552423141529
// MI455X (gfx1250) — compile-verified
//
#include <hip/hip_runtime.h>

// GCN encoder (VGAE) for MI455X / gfx1250.
// Dense GEMMs use V_WMMA_F32_16X16X4_F32 (full f32 precision; GEMM FLOPs are
// ~1.2 GFLOP total, negligible vs the edge aggregation). Each wave computes
// MT row-tiles with NO in-loop guards (remainder handled by a separate 1-tile
// launch), so every B fragment load feeds MT WMMAs in straight-line code.
// Edge aggregation uses float4 gathers + global_atomic_add_f32; all working
// sets (<70MB) fit in the 192MB L2, so gathers and atomics resolve at L2
// bandwidth — the true bottleneck for this workload.

typedef __attribute__((ext_vector_type(2))) float v2f;
typedef __attribute__((ext_vector_type(8))) float v8f;

#define N_NODES 50000
#define N_EDGES 1600000
#define DIM_IN  128
#define DIM_HID 64
#define DIM_OUT 32

// 3125 row tiles = 781 blocks * 4 tiles + 1 tail tile
#define MT_MAIN   4
#define MAIN_BLKS 781
#define TAIL_ROW0 (MAIN_BLKS * 16 * MT_MAIN)   // 49984

// ---------------- degree / norm ----------------

__global__ void k_init_deg(float* __restrict__ deg, int n) {
    int i = blockIdx.x * blockDim.x + threadIdx.x;
    if (i < n) deg[i] = 1.0f;                       // self-loop
}

__global__ void k_degree(const long long* __restrict__ dst,
                         float* __restrict__ deg, int e) {
    int i = blockIdx.x * blockDim.x + threadIdx.x;
    if (i < e) atomicAdd(&deg[(int)dst[i]], 1.0f);
}

__global__ void k_rsqrt(float* __restrict__ deg, int n) {
    int i = blockIdx.x * blockDim.x + threadIdx.x;
    if (i < n) deg[i] = rsqrtf(deg[i]);
}

// ---------------- WMMA f32 GEMM: C = A[M,K] @ B[K,Nc] ----------------------
// Block = (Nc/16) waves; wave w owns column tile w for MT consecutive row
// tiles starting at row0 + blockIdx*16*MT. No bounds checks: the grid covers
// exactly full tiles (tail tile launched separately with MT=1). B fragments
// are loaded once per k-step and reused by all MT WMMAs. Fragment layouts per
// CDNA5 ISA 7.12.2 (32-bit A 16x4; 32-bit C/D 16x16), wave32.

template <int MT>
__global__ __launch_bounds__(128)
void k_gemm_wmma(const float* __restrict__ A, const float* __restrict__ B,
                 float* __restrict__ C, int row0, int K, int Nc) {
    const int wave = threadIdx.x >> 5;
    const int lane = threadIdx.x & 31;
    if (wave >= (Nc >> 4)) return;

    const int nbase  = wave << 4;
    const int rc     = lane & 15;   // A row / B-C column within tile
    const int hi     = lane >> 4;   // half-wave select
    const int mbase0 = row0 + blockIdx.x * (16 * MT);

    const float* __restrict__ arow[MT];
#pragma unroll
    for (int t = 0; t < MT; ++t)
        arow[t] = A + (size_t)(mbase0 + 16 * t + rc) * K;

    v8f acc[MT];
#pragma unroll
    for (int t = 0; t < MT; ++t) acc[t] = (v8f){};

    for (int k0 = 0; k0 < K; k0 += 4) {
        const int ka = k0 + 2 * hi;
        v2f b; b.x = B[(size_t)ka       * Nc + nbase + rc];
               b.y = B[(size_t)(ka + 1) * Nc + nbase + rc];
#pragma unroll
        for (int t = 0; t < MT; ++t) {
            v2f a; a.x = arow[t][ka];
                   a.y = arow[t][ka + 1];
            // D = A*B + C (8 args: neg_a,A,neg_b,B,c_mod,C,reuse_a,reuse_b)
            acc[t] = __builtin_amdgcn_wmma_f32_16x16x4_f32(
                false, a, false, b, (short)0, acc[t], false, false);
        }
    }

#pragma unroll
    for (int t = 0; t < MT; ++t) {
#pragma unroll
        for (int v = 0; v < 8; ++v) {
            C[(size_t)(mbase0 + 16 * t + v + 8 * hi) * Nc + nbase + rc] =
                acc[t][v];
        }
    }
}

// ------------- self-loop + bias: out = sup * dinv^2 + bias -------------

template <int F>
__global__ void k_selfloop_bias(const float* __restrict__ sup,
                                const float* __restrict__ dinv,
                                const float* __restrict__ bias,
                                float* __restrict__ out, int n) {
    int i = blockIdx.x * blockDim.x + threadIdx.x;
    if (i >= n * F) return;
    int node = i / F;          // F is a power of two -> shift
    int j    = i & (F - 1);
    float d = dinv[node];
    out[i] = sup[i] * d * d + bias[j];
}

// ------------- edge scatter: out[dst] += sup[src] * dinv[src]*dinv[dst] ----
// One thread handles 4 contiguous features of one edge (float4 gather,
// 4x global_atomic_add_f32 scatter). All resident in L2.

template <int F>
__global__ void k_scatter(const long long* __restrict__ src,
                          const long long* __restrict__ dst,
                          const float* __restrict__ dinv,
                          const float* __restrict__ sup,
                          float* __restrict__ out, int e) {
    constexpr int FQ = F / 4;              // float4 groups per edge
    const int idx  = blockIdx.x * blockDim.x + threadIdx.x;
    const int edge = idx / FQ;             // power of two -> shift
    const int g    = idx & (FQ - 1);
    if (edge >= e) return;

    const int s = (int)src[edge];
    const int d = (int)dst[edge];
    const float norm = dinv[s] * dinv[d];

    const float4 v = ((const float4*)(sup + (size_t)s * F))[g];
    float* op = out + (size_t)d * F + (g << 2);
    atomicAdd(op + 0, v.x * norm);
    atomicAdd(op + 1, v.y * norm);
    atomicAdd(op + 2, v.z * norm);
    atomicAdd(op + 3, v.w * norm);
}

// ------------- clamped leaky relu: clip(leaky(x,0.01), -3, 1) -------------

__global__ void k_act(const float* __restrict__ in, float* __restrict__ out,
                      int n) {
    int i = blockIdx.x * blockDim.x + threadIdx.x;
    if (i >= n) return;
    float x = in[i];
    float y = x >= 0.0f ? x : 0.01f * x;
    out[i] = fminf(fmaxf(y, -3.0f), 1.0f);
}

// ---------------------------------------------------------------------------

static inline void launch_gemm(const float* A, const float* B, float* C,
                               int K, int Nc, hipStream_t stream) {
    // main: 781 blocks x 4 tiles (rows 0..49983), tail: 1 block x 1 tile
    k_gemm_wmma<MT_MAIN><<<MAIN_BLKS, 32 * (Nc / 16), 0, stream>>>(
        A, B, C, 0, K, Nc);
    k_gemm_wmma<1><<<1, 32 * (Nc / 16), 0, stream>>>(
        A, B, C, TAIL_ROW0, K, Nc);
}

extern "C" void kernel_launch(void* const* d_in, const int* in_sizes, int n_in,
                              void* d_out, int out_size, void* d_ws, size_t ws_size,
                              hipStream_t stream) {
    (void)in_sizes; (void)n_in; (void)out_size; (void)ws_size;

    const float*     x    = (const float*)d_in[0];
    const long long* edge = (const long long*)d_in[1];   // int64 (2,E)
    const float*     W1   = (const float*)d_in[2];
    const float*     b1   = (const float*)d_in[3];
    const float*     Wm   = (const float*)d_in[4];
    const float*     bm   = (const float*)d_in[5];
    const float*     Ws   = (const float*)d_in[6];
    const float*     bs   = (const float*)d_in[7];

    const long long* srcI = edge;
    const long long* dstI = edge + N_EDGES;

    // Output layout (flat, return order):
    float* out        = (float*)d_out;
    float* h          = out;                                     // N*64
    float* mean       = h        + (size_t)N_NODES * DIM_HID;    // N*32
    float* logstd     = mean     + (size_t)N_NODES * DIM_OUT;    // N*32
    float* x_emp      = logstd   + (size_t)N_NODES * DIM_OUT;    // N*64
    float* mean_emp   = x_emp    + (size_t)N_NODES * DIM_HID;    // N*32
    float* logstd_emp = mean_emp + (size_t)N_NODES * DIM_OUT;    // N*32

    // Workspace: dinv [N] | sup1 [N*64] | supM [N*32] | supS [N*32]
    float* ws   = (float*)d_ws;
    float* dinv = ws;
    float* sup1 = dinv + N_NODES;
    float* supM = sup1 + (size_t)N_NODES * DIM_HID;
    float* supS = supM + (size_t)N_NODES * DIM_OUT;

    const int T = 256;

    // degrees -> d_inv_sqrt
    k_init_deg<<<(N_NODES + T - 1) / T, T, 0, stream>>>(dinv, N_NODES);
    k_degree  <<<(N_EDGES + T - 1) / T, T, 0, stream>>>(dstI, dinv, N_EDGES);
    k_rsqrt   <<<(N_NODES + T - 1) / T, T, 0, stream>>>(dinv, N_NODES);

    // layer 1: sup1 = x @ W1 ; x_emp = Ahat*sup1 + b1
    launch_gemm(x, W1, sup1, DIM_IN, DIM_HID, stream);
    k_selfloop_bias<DIM_HID><<<(N_NODES * DIM_HID + T - 1) / T, T, 0, stream>>>(
        sup1, dinv, b1, x_emp, N_NODES);
    k_scatter<DIM_HID><<<(N_EDGES * (DIM_HID / 4) + T - 1) / T, T, 0, stream>>>(
        srcI, dstI, dinv, sup1, x_emp, N_EDGES);
    k_act<<<(N_NODES * DIM_HID + T - 1) / T, T, 0, stream>>>(
        x_emp, h, N_NODES * DIM_HID);

    // layer 2 (mean): supM = h @ Wm ; mean_emp = Ahat*supM + bm
    launch_gemm(h, Wm, supM, DIM_HID, DIM_OUT, stream);
    k_selfloop_bias<DIM_OUT><<<(N_NODES * DIM_OUT + T - 1) / T, T, 0, stream>>>(
        supM, dinv, bm, mean_emp, N_NODES);
    k_scatter<DIM_OUT><<<(N_EDGES * (DIM_OUT / 4) + T - 1) / T, T, 0, stream>>>(
        srcI, dstI, dinv, supM, mean_emp, N_EDGES);

    // layer 2 (logstd): supS = h @ Ws ; logstd_emp = Ahat*supS + bs
    launch_gemm(h, Ws, supS, DIM_HID, DIM_OUT, stream);
    k_selfloop_bias<DIM_OUT><<<(N_NODES * DIM_OUT + T - 1) / T, T, 0, stream>>>(
        supS, dinv, bs, logstd_emp, N_NODES);
    k_scatter<DIM_OUT><<<(N_EDGES * (DIM_OUT / 4) + T - 1) / T, T, 0, stream>>>(
        srcI, dstI, dinv, supS, logstd_emp, N_EDGES);

    // mean_emp & logstd_emp are contiguous, as are mean & logstd:
    // one fused activation pass over both.
    k_act<<<(2 * N_NODES * DIM_OUT + T - 1) / T, T, 0, stream>>>(
        mean_emp, mean, 2 * N_NODES * DIM_OUT);
}